// ProposalLayer_52372831207791
// MI455X (gfx1250) — compile-verified
//
#include <hip/hip_runtime.h>
#include <math.h>

#define IMG_SIZE   800.0f
#define NMS_THRESH 0.7f
#define N_PRE      2000
#define N_POST     1000
#define MIN_SIZE   16.0f
#define NBINS      4096
#define N_CAND     8192
#define MASK_WORDS 64        // 2000 bits -> 63 words, pad to 64
#define N_TILES    125       // 2000 / 16

typedef __attribute__((ext_vector_type(2))) float v2f;
typedef __attribute__((ext_vector_type(8))) float v8f;

// ---------------- shared decode helper (matches reference math) --------------
__device__ __forceinline__ float4 decode_clip(const float4 a, const float4 l) {
    float w  = a.z - a.x;
    float h  = a.w - a.y;
    float cx = a.x + 0.5f * w;
    float cy = a.y + 0.5f * h;
    float pcx = cx + l.x * w;
    float pcy = cy + l.y * h;
    float pw  = w * expf(l.z);
    float ph  = h * expf(l.w);
    float x1 = pcx - 0.5f * pw, y1 = pcy - 0.5f * ph;
    float x2 = pcx + 0.5f * pw, y2 = pcy + 0.5f * ph;
    x1 = fminf(fmaxf(x1, 0.0f), IMG_SIZE);
    y1 = fminf(fmaxf(y1, 0.0f), IMG_SIZE);
    x2 = fminf(fmaxf(x2, 0.0f), IMG_SIZE);
    y2 = fminf(fmaxf(y2, 0.0f), IMG_SIZE);
    return make_float4(x1, y1, x2, y2);
}

// ---------------- 0: zero scratch (hist+misc+mask contiguous) + output -------
__global__ void k_zero(unsigned* zws, int nz, float* out, int nout) {
    int i = blockIdx.x * blockDim.x + threadIdx.x;
    if (i < nz)   zws[i] = 0u;
    if (i < nout) out[i] = 0.0f;
}

// ---------------- 1: streaming decode + mask + score + histogram -------------
__global__ void k_decode_hist(const float4* __restrict__ locs,
                              const float*  __restrict__ scores,
                              const float4* __restrict__ anchors,
                              float* __restrict__ mscore,
                              unsigned* __restrict__ hist, int n) {
    __shared__ unsigned lh[NBINS];
    for (int i = threadIdx.x; i < NBINS; i += blockDim.x) lh[i] = 0u;
    __syncthreads();
    for (int i = blockIdx.x * blockDim.x + threadIdx.x; i < n;
         i += gridDim.x * blockDim.x) {
        float4 roi = decode_clip(anchors[i], locs[i]);
        float ws = roi.z - roi.x, hs = roi.w - roi.y;
        bool valid = (ws >= MIN_SIZE) && (hs >= MIN_SIZE);
        float s = valid ? scores[i] : -INFINITY;
        mscore[i] = s;
        if (valid) {
            int b = (int)(s * (float)NBINS);
            b = b < 0 ? 0 : (b > NBINS - 1 ? NBINS - 1 : b);
            atomicAdd(&lh[b], 1u);
        }
    }
    __syncthreads();
    for (int i = threadIdx.x; i < NBINS; i += blockDim.x)
        if (lh[i]) atomicAdd(&hist[i], lh[i]);
}

// ---------------- 2: find threshold bin (suffix cumsum crosses N_PRE) --------
__global__ void k_thresh(const unsigned* __restrict__ hist, unsigned* misc) {
    if (threadIdx.x == 0) {
        unsigned acc = 0; int T = 0;
        for (int b = NBINS - 1; b >= 0; --b) {
            acc += hist[b];
            if (acc >= (unsigned)N_PRE) { T = b; break; }
        }
        misc[1] = (unsigned)T;
        ((float*)misc)[2] = (float)T / (float)NBINS;  // exact: T / 2^12
    }
}

// ---------------- 3: compact candidates above threshold ----------------------
__global__ void k_compact(const float* __restrict__ mscore, unsigned* misc,
                          float* __restrict__ cScore, int* __restrict__ cIdx,
                          int n) {
    float binLo = ((const float*)misc)[2];
    for (int i = blockIdx.x * blockDim.x + threadIdx.x; i < n;
         i += gridDim.x * blockDim.x) {
        float s = mscore[i];
        if (s >= binLo) {
            unsigned p = atomicAdd(&misc[0], 1u);
            if (p < (unsigned)N_CAND) { cScore[p] = s; cIdx[p] = i; }
        }
    }
}

// ---------------- 4: one-block bitonic sort (score desc, idx asc) ------------
__global__ __launch_bounds__(1024)
void k_sort(const float* __restrict__ cScore, const int* __restrict__ cIdx,
            const unsigned* misc, int* __restrict__ selIdx) {
    __shared__ float ss[N_CAND];
    __shared__ int   si[N_CAND];
    unsigned cnt = misc[0];
    if (cnt > (unsigned)N_CAND) cnt = N_CAND;
    for (int t = threadIdx.x; t < N_CAND; t += blockDim.x) {
        if (t < (int)cnt) { ss[t] = cScore[t]; si[t] = cIdx[t]; }
        else              { ss[t] = -INFINITY; si[t] = 0x7FFFFFFF; }
    }
    __syncthreads();
    for (int k = 2; k <= N_CAND; k <<= 1) {
        for (int j = k >> 1; j > 0; j >>= 1) {
            for (int e = threadIdx.x; e < N_CAND; e += blockDim.x) {
                int x = e ^ j;
                if (x > e) {
                    float s1 = ss[e], s2 = ss[x];
                    int   i1 = si[e], i2 = si[x];
                    bool up = ((e & k) == 0);
                    // comesFirst(a,b): a.s > b.s || (== && a.i < b.i)
                    bool sw = up ? ((s2 > s1) || (s2 == s1 && i2 < i1))
                                 : ((s1 > s2) || (s1 == s2 && i1 < i2));
                    if (sw) { ss[e] = s2; ss[x] = s1; si[e] = i2; si[x] = i1; }
                }
            }
            __syncthreads();
        }
    }
    for (int t = threadIdx.x; t < N_PRE; t += blockDim.x)
        selIdx[t] = (ss[t] >= 0.0f) ? si[t] : -1;
}

// ---------------- 5: gather + re-decode the 2000 selected boxes --------------
__global__ void k_gather(const int* __restrict__ selIdx,
                         const float4* __restrict__ locs,
                         const float4* __restrict__ anchors,
                         float4* __restrict__ roiK, float* __restrict__ areaK,
                         int* __restrict__ validK) {
    int k = blockIdx.x * blockDim.x + threadIdx.x;
    if (k >= N_PRE) return;
    int i = selIdx[k];
    if (i >= 0) {
        float4 roi = decode_clip(anchors[i], locs[i]);
        roiK[k]  = roi;
        areaK[k] = (roi.z - roi.x) * (roi.w - roi.y);
        validK[k] = 1;
    } else {
        roiK[k]  = make_float4(0.f, 0.f, 0.f, 0.f);
        areaK[k] = 0.f;
        validK[k] = 0;
    }
}

// ---------------- 6: suppression bit-matrix; WMMA computes area_i + area_j ---
__global__ __launch_bounds__(32)
void k_mask(const float4* __restrict__ roiK, const float* __restrict__ areaK,
            unsigned* __restrict__ mask) {
    int tI = blockIdx.y, tJ = blockIdx.x;
    if (tJ < tI) return;  // only j > i can suppress (wave-uniform branch)
    __shared__ float4 bi[16], bj[16];
    __shared__ float  ai[16], aj[16];
    int lane = threadIdx.x;
    if (lane < 16) { bi[lane] = roiK[tI * 16 + lane]; ai[lane] = areaK[tI * 16 + lane]; }
    else { int l = lane - 16; bj[l] = roiK[tJ * 16 + l]; aj[l] = areaK[tJ * 16 + l]; }
    __syncthreads();

    bool lo = lane < 16;
    int  ln = lane & 15;
    // Rank-2 outer sum via V_WMMA_F32_16X16X4_F32:
    //   A(16x4): col0 = area_i, col1 = 1;  B(4x16): row0 = 1, row1 = area_j
    //   D[m][n] = area_m + area_n  (A/B share the slot->K mapping, so this is
    //   correct under any consistent hardware K ordering)
    v2f A; A.x = lo ? ai[ln] : 0.0f; A.y = lo ? 1.0f   : 0.0f;
    v2f B; B.x = lo ? 1.0f   : 0.0f; B.y = lo ? aj[ln] : 0.0f;
    v8f C = {};
    C = __builtin_amdgcn_wmma_f32_16x16x4_f32(false, A, false, B,
                                              (short)0, C, false, false);

    unsigned wordIdx = (unsigned)(tJ >> 1);
    unsigned shift   = (unsigned)(tJ & 1) * 16u;
    float4 bbj = bj[ln];
    for (int r = 0; r < 8; ++r) {
        int m  = lo ? r : r + 8;              // C/D layout: VGPR r, lane-half
        int ig = tI * 16 + m;
        int jg = tJ * 16 + ln;
        float4 bbi = bi[m];
        float ix1 = fmaxf(bbi.x, bbj.x);
        float iy1 = fmaxf(bbi.y, bbj.y);
        float ix2 = fminf(bbi.z, bbj.z);
        float iy2 = fminf(bbi.w, bbj.w);
        float inter = fmaxf(ix2 - ix1, 0.0f) * fmaxf(iy2 - iy1, 0.0f);
        float uni   = C[r] - inter;           // (area_i + area_j) - inter
        // iou > t  <=>  inter > t * max(union, 1e-9)   (divide-free)
        bool pred   = (inter > NMS_THRESH * fmaxf(uni, 1e-9f)) && (jg > ig);
        unsigned b32 = (unsigned)__ballot(pred);
        if (lane == 0) {
            unsigned blo = (b32 & 0xFFFFu) << shift;         // rows r
            unsigned bhi = ((b32 >> 16) & 0xFFFFu) << shift; // rows r+8
            if (blo) atomicOr(&mask[(tI * 16 + r) * MASK_WORDS + wordIdx], blo);
            if (bhi) atomicOr(&mask[(tI * 16 + r + 8) * MASK_WORDS + wordIdx], bhi);
        }
    }
}

// ---------------- 7: single-wave greedy NMS scan (register-resident keep) ----
__global__ __launch_bounds__(32)
void k_scan(const int* __restrict__ validK, const unsigned* __restrict__ mask,
            unsigned* __restrict__ keepW) {
    int lane = threadIdx.x;
    unsigned ka = 0u, kb = 0u;
    for (int b = 0; b < 32; ++b) {
        int e0 = lane * 32 + b;
        int e1 = (lane + 32) * 32 + b;
        if (e0 < N_PRE && validK[e0]) ka |= (1u << b);
        if (e1 < N_PRE && validK[e1]) kb |= (1u << b);
    }
    for (int i = 0; i < N_PRE; ++i) {
        int wi = i >> 5;
        unsigned wlo = __shfl(ka, wi & 31, 32);
        unsigned whi = __shfl(kb, wi & 31, 32);
        unsigned w = (wi < 32) ? wlo : whi;
        if ((w >> (i & 31)) & 1u) {
            ka &= ~mask[i * MASK_WORDS + lane];
            kb &= ~mask[i * MASK_WORDS + 32 + lane];
        }
    }
    keepW[lane]      = ka;
    keepW[lane + 32] = kb;
}

// ---------------- 8: stable-compact kept boxes into output -------------------
__global__ __launch_bounds__(1024)
void k_final(const unsigned* __restrict__ keepW, const float4* __restrict__ roiK,
             float4* __restrict__ out) {
    __shared__ unsigned wpre[MASK_WORDS];
    __shared__ unsigned wcnt[MASK_WORDS];
    if (threadIdx.x < MASK_WORDS) wcnt[threadIdx.x] = __popc(keepW[threadIdx.x]);
    __syncthreads();
    if (threadIdx.x == 0) {
        unsigned acc = 0;
        for (int w = 0; w < MASK_WORDS; ++w) { wpre[w] = acc; acc += wcnt[w]; }
    }
    __syncthreads();
    for (int e = threadIdx.x; e < N_PRE; e += blockDim.x) {
        int w = e >> 5, bp = e & 31;
        unsigned word = keepW[w];
        if ((word >> bp) & 1u) {
            unsigned pos = wpre[w] + __popc(word & ((1u << bp) - 1u));
            if (pos < (unsigned)N_POST) out[pos] = roiK[e];
        }
    }
}

// ---------------- host: orchestrate pipeline on `stream` --------------------
extern "C" void kernel_launch(void* const* d_in, const int* in_sizes, int n_in,
                              void* d_out, int out_size, void* d_ws, size_t ws_size,
                              hipStream_t stream) {
    const float4* locs    = (const float4*)d_in[0];
    const float*  scores  = (const float*) d_in[1];
    const float4* anchors = (const float4*)d_in[2];
    float* out = (float*)d_out;
    const int N = in_sizes[1];               // number of anchors

    // workspace layout (byte offsets)
    char* base = (char*)d_ws;
    size_t off = 0;
    float*    mscore = (float*)(base + off);            off += (size_t)N * 4;
    size_t zbeg = off;                                  // zero region start
    unsigned* hist   = (unsigned*)(base + off);         off += NBINS * 4;
    unsigned* misc   = (unsigned*)(base + off);         off += 256;
    unsigned* mask   = (unsigned*)(base + off);         off += (size_t)N_PRE * MASK_WORDS * 4;
    size_t zlen32 = (off - zbeg) / 4;                   // hist+misc+mask
    float*    cScore = (float*)(base + off);            off += N_CAND * 4;
    int*      cIdx   = (int*)(base + off);              off += N_CAND * 4;
    int*      selIdx = (int*)(base + off);              off += 2048 * 4;
    float4*   roiK   = (float4*)(base + off);           off += (size_t)N_PRE * 16;
    float*    areaK  = (float*)(base + off);            off += 2048 * 4;
    int*      validK = (int*)(base + off);              off += 2048 * 4;
    unsigned* keepW  = (unsigned*)(base + off);         off += MASK_WORDS * 4;

    // 0: zero hist/misc/mask + output
    {
        int nz = (int)zlen32;
        int mx = nz > out_size ? nz : out_size;
        k_zero<<<(mx + 255) / 256, 256, 0, stream>>>((unsigned*)(base + zbeg), nz,
                                                     out, out_size);
    }
    // 1: decode + mask + score + histogram (bandwidth-bound streaming pass)
    k_decode_hist<<<2048, 256, 0, stream>>>(locs, scores, anchors, mscore, hist, N);
    // 2: threshold bin
    k_thresh<<<1, 32, 0, stream>>>(hist, misc);
    // 3: candidate compaction
    k_compact<<<2048, 256, 0, stream>>>(mscore, misc, cScore, cIdx, N);
    // 4: exact top-2000 via one-block bitonic sort
    k_sort<<<1, 1024, 0, stream>>>(cScore, cIdx, misc, selIdx);
    // 5: gather + re-decode selected boxes
    k_gather<<<(N_PRE + 255) / 256, 256, 0, stream>>>(selIdx, locs, anchors,
                                                      roiK, areaK, validK);
    // 6: suppression bit-matrix (WMMA outer-sum + ballot bitpack)
    k_mask<<<dim3(N_TILES, N_TILES), 32, 0, stream>>>(roiK, areaK, mask);
    // 7: serial greedy NMS, single wave
    k_scan<<<1, 32, 0, stream>>>(validK, mask, keepW);
    // 8: compact kept boxes to output
    k_final<<<1, 1024, 0, stream>>>(keepW, roiK, (float4*)out);
}